// RPNLayer_85383949844583
// MI455X (gfx1250) — compile-verified
//
#include <hip/hip_runtime.h>
#include <hip/hip_bf16.h>

typedef __attribute__((ext_vector_type(2))) float v2f;
typedef __attribute__((ext_vector_type(8))) float v8f;

#define BB 8
#define SS 4096
#define DD 1024
#define AA 3
#define CC 2
#define KK 64
#define NPOS (SS * AA)   // 12288 candidate positions per batch
#define MTOT (BB * SS)   // 32768 GEMM rows
#define NPAD 16          // padded N for WMMA

__device__ __forceinline__ unsigned long long umax64(unsigned long long a,
                                                     unsigned long long b) {
    return a > b ? a : b;
}

// ---------------------------------------------------------------------------
// Kernel 0: zero the loss accumulator (graph replays must be deterministic)
// ---------------------------------------------------------------------------
__global__ void init_kernel(float* out) {
    if (threadIdx.x == 0) out[0] = 0.0f;
}

// ---------------------------------------------------------------------------
// Kernel 0b: zero-pad W (1024x6) -> Wpad (1024x16) so B-fragment loads in the
// GEMM are unconditional (kills the exec-mask branches seen in round 1).
// ---------------------------------------------------------------------------
__global__ void pad_w_kernel(const float* __restrict__ W,
                             float* __restrict__ Wpad) {
    int i = blockIdx.x * blockDim.x + threadIdx.x;  // 0..16383
    if (i < DD * NPAD) {
        int k = i >> 4;
        int n = i & 15;
        Wpad[i] = (n < AA * CC) ? W[k * (AA * CC) + n] : 0.0f;
    }
}

// ---------------------------------------------------------------------------
// Kernel 1: logits = X(32768x1024) @ W(1024x6) + b  via V_WMMA_F32_16X16X4_F32
// One wave per 16-row M-tile; K in LDS chunks of 64 staged with b128 ops.
// LDS row stride 68: 16B-aligned float4 stores, and 68 mod 64 = 4 -> the 16
// fragment lanes (m=0..15) hit 16 distinct banks. Prefetch next K-chunk.
// ---------------------------------------------------------------------------
__global__ __launch_bounds__(128) void gemm_logits_kernel(
    const float* __restrict__ X, const float* __restrict__ Wpad,
    const float* __restrict__ bias, float* __restrict__ logits) {
    __shared__ float ldsA[4][16 * 68];

    const int wave = threadIdx.x >> 5;
    const int lane = threadIdx.x & 31;
    const int tile = blockIdx.x * 4 + wave;  // 0..2047
    const int row0 = tile * 16;
    const int n = lane & 15;                 // N index (B/C/D frag)
    const int m = lane & 15;                 // M index (A frag)
    const int kbase = (lane >> 4) * 2;       // lanes 0-15: K+0/1, 16-31: K+2/3

    v8f acc = {};
    float* lA = &ldsA[wave][0];
    const float* Xrow = X + (size_t)row0 * DD;

    for (int k0 = 0; k0 < DD; k0 += 64) {
        // stage 16 rows x 64 cols, float4 per lane (coalesced b128)
#pragma unroll
        for (int i = 0; i < 8; ++i) {
            int idx = (i * 32 + lane) * 4;  // element 0..1023
            int r = idx >> 6;
            int c = idx & 63;
            float4 v = *(const float4*)(Xrow + (size_t)r * DD + k0 + c);
            *(float4*)(lA + r * 68 + c) = v;
        }
        __syncthreads();

        if (k0 + 64 < DD) {  // prefetch next chunk (global_prefetch_b8)
            int pr = lane >> 1;
            int pc = (lane & 1) * 32;
            __builtin_prefetch(Xrow + (size_t)pr * DD + k0 + 64 + pc, 0, 3);
        }

#pragma unroll
        for (int kk = 0; kk < 64; kk += 4) {
            v2f a, b;
            a.x = lA[m * 68 + kk + kbase];
            a.y = lA[m * 68 + kk + kbase + 1];
            int kg = k0 + kk + kbase;
            b.x = Wpad[kg * NPAD + n];
            b.y = Wpad[(kg + 1) * NPAD + n];
            acc = __builtin_amdgcn_wmma_f32_16x16x4_f32(
                false, a, false, b, (short)0, acc, false, false);
        }
        __syncthreads();
    }

    // C/D layout: VGPR r, lanes 0-15 -> M=r, lanes 16-31 -> M=r+8; N = lane&15
    if (n < AA * CC) {
        float bb = bias[n];
        int mhalf = (lane >> 4) * 8;
#pragma unroll
        for (int r = 0; r < 8; ++r) {
            int mm = row0 + mhalf + r;
            logits[(size_t)mm * (AA * CC) + n] = acc[r] + bb;
        }
    }
}

// ---------------------------------------------------------------------------
// Kernel 2: predict_label = argmax over C=2 (first-max tie-break -> strict >)
// ---------------------------------------------------------------------------
__global__ void predict_kernel(const float* __restrict__ logits,
                               float* __restrict__ out_pred, int total) {
    int i = blockIdx.x * blockDim.x + threadIdx.x;
    if (i < total) {
        float x0 = logits[2 * i];
        float x1 = logits[2 * i + 1];
        out_pred[i] = (x1 > x0) ? 1.0f : 0.0f;
    }
}

// ---------------------------------------------------------------------------
// Kernel 3: per-batch stable top-64 + gathers + margin loss.
// Fast path (>=64 positives, which is certain for Bernoulli(0.5) x 12288):
//   positives are forced to 2.0 > sigmoid(x), and top_k tie-breaks ascending
//   index, so the winners are exactly the first 64 positive flat indices.
//   One label pass + block prefix scan.
// Slow path (<64 positives): exact stable 64-round argmax over u64 keys in
//   96KB dynamic LDS (value bits desc, ~index asc).
// ---------------------------------------------------------------------------
__global__ __launch_bounds__(1024) void topk_loss_kernel(
    const float* __restrict__ logits, const int* __restrict__ labels,
    float* __restrict__ out) {
    extern __shared__ unsigned long long keys[];  // NPOS (slow path only)
    __shared__ int scan[1024];
    __shared__ unsigned long long red[1024];
    __shared__ int sel[KK];
    __shared__ float lsum[KK];

    const int b = blockIdx.x;
    const int tid = threadIdx.x;
    const int* labB = labels + b * NPOS;  // (b*S+s)*A + a == b*NPOS + p
    const int PER = NPOS / 1024;          // 12 contiguous positions per thread
    const int base = tid * PER;

    int loc[PER];
    int cnt = 0;
#pragma unroll
    for (int j = 0; j < PER; ++j) {
        loc[j] = labB[base + j];
        cnt += (loc[j] == 1);
    }
    scan[tid] = cnt;
    __syncthreads();
    // Hillis-Steele inclusive scan over 1024 threads
    for (int off = 1; off < 1024; off <<= 1) {
        int t = (tid >= off) ? scan[tid - off] : 0;
        __syncthreads();
        scan[tid] += t;
        __syncthreads();
    }
    int total_pos = scan[1023];
    int excl = scan[tid] - cnt;

    if (total_pos >= KK) {
        // fast path: first KK positive indices, ascending
        if (excl < KK && cnt > 0) {
            int r = excl;
#pragma unroll
            for (int j = 0; j < PER; ++j) {
                if (loc[j] == 1) {
                    if (r < KK) sel[r] = base + j;
                    ++r;
                }
            }
        }
        __syncthreads();
    } else {
        // exact slow path (never taken in practice)
        for (int p = tid; p < NPOS; p += 1024) {
            int s = p / AA;
            int a = p - AA * s;
            int lab = labB[p];
            float x1 = logits[((size_t)(b * SS + s)) * (AA * CC) + a * CC + 1];
            float v = (lab == 1) ? 2.0f : (1.0f / (1.0f + __expf(-x1)));
            keys[p] = ((unsigned long long)__float_as_uint(v) << 32) |
                      (unsigned long long)(0xFFFFFFFFu - (unsigned)p);
        }
        __syncthreads();
        for (int k = 0; k < KK; ++k) {
            unsigned long long best = 0ull;
            for (int p = tid; p < NPOS; p += 1024) best = umax64(best, keys[p]);
            red[tid] = best;
            __syncthreads();
#pragma unroll
            for (int off = 512; off > 0; off >>= 1) {
                if (tid < off) red[tid] = umax64(red[tid], red[tid + off]);
                __syncthreads();
            }
            if (tid == 0) {
                unsigned int p =
                    0xFFFFFFFFu - (unsigned)(red[0] & 0xFFFFFFFFull);
                sel[k] = (int)p;
                keys[p] = 0ull;
            }
            __syncthreads();
        }
    }

    // Output layout (all f32): [0]=loss, [1..)=predict, total_idx, cand_label
    const int IDX_OFF = 1 + BB * SS * AA;
    const int CAND_OFF = IDX_OFF + BB * KK * 3;

    if (tid < KK) {
        int p = sel[tid];
        int s = p / AA;
        int a = p - AA * s;
        int row = b * KK + tid;
        out[IDX_OFF + row * 3 + 0] = (float)b;
        out[IDX_OFF + row * 3 + 1] = (float)s;
        out[IDX_OFF + row * 3 + 2] = (float)a;

        size_t lbase = ((size_t)(b * SS + s)) * (AA * CC) + a * CC;
        float x0 = logits[lbase];
        float x1 = logits[lbase + 1];
        out[CAND_OFF + row] = (x1 > x0) ? 1.0f : 0.0f;

        int y = labB[p];
        float xy = (y == 1) ? x1 : x0;
        float xo = (y == 1) ? x0 : x1;
        lsum[tid] = fmaxf(0.0f, 5.0f - xy + xo) * 0.5f;  // /C, C=2
    }
    __syncthreads();
    if (tid == 0) {
        float s = 0.0f;
#pragma unroll
        for (int i = 0; i < KK; ++i) s += lsum[i];
        atomicAdd(&out[0], s / (float)(BB * KK));  // mean over 512 samples
    }
}

// ---------------------------------------------------------------------------
extern "C" void kernel_launch(void* const* d_in, const int* in_sizes, int n_in,
                              void* d_out, int out_size, void* d_ws,
                              size_t ws_size, hipStream_t stream) {
    const float* X = (const float*)d_in[0];     // (B,S,D) f32
    const float* W = (const float*)d_in[1];     // (D, A*C) f32
    const float* bias = (const float*)d_in[2];  // (A*C,) f32
    const int* labels = (const int*)d_in[3];    // (B,S,A) int32
    float* out = (float*)d_out;

    float* logits = (float*)d_ws;               // 196608 floats
    float* Wpad = logits + (size_t)MTOT * (AA * CC);  // 16384 floats

    init_kernel<<<1, 32, 0, stream>>>(out);
    pad_w_kernel<<<(DD * NPAD + 255) / 256, 256, 0, stream>>>(W, Wpad);

    gemm_logits_kernel<<<512, 128, 0, stream>>>(X, Wpad, bias, logits);

    int total = BB * SS * AA;
    predict_kernel<<<(total + 255) / 256, 256, 0, stream>>>(logits, out + 1,
                                                            total);

    topk_loss_kernel<<<BB, 1024, NPOS * sizeof(unsigned long long), stream>>>(
        logits, labels, out);
}